// GNN_87316685128564
// MI455X (gfx1250) — compile-verified
//
#include <hip/hip_runtime.h>

#define N_NODES 8192
#define N_EDGES 131072
#define F_IN 4
#define H1 32
#define H2 16

typedef __attribute__((ext_vector_type(2))) float v2f;
typedef __attribute__((ext_vector_type(8))) float v8f;

// ---- degree / norm ---------------------------------------------------------
__global__ void k_deg_init(float* __restrict__ deg) {
  int i = blockIdx.x * blockDim.x + threadIdx.x;
  if (i < N_NODES) deg[i] = 1.0f;  // self-loop contributes 1
}

__global__ void k_deg_count(const int* __restrict__ dst, float* __restrict__ deg) {
  int e = blockIdx.x * blockDim.x + threadIdx.x;
  if (e < N_EDGES) atomicAdd(&deg[dst[e]], 1.0f);
}

__global__ void k_deg_rsqrt(float* __restrict__ dinv) {
  int i = blockIdx.x * blockDim.x + threadIdx.x;
  if (i < N_NODES) dinv[i] = rsqrtf(dinv[i]);  // deg >= 1 always
}

// ---- aggregation: agg = Â h  (self-loop init + edge scatter) ---------------
template <int F>
__global__ void k_agg_init(const float* __restrict__ h, const float* __restrict__ dinv,
                           float* __restrict__ agg) {
  int t = blockIdx.x * blockDim.x + threadIdx.x;
  if (t < N_NODES * F) {
    int i = t / F;
    float di = dinv[i];
    agg[t] = h[t] * di * di;  // self-loop: norm = dinv[i]^2
  }
}

template <int F>
__global__ void k_edge_scatter(const int* __restrict__ src, const int* __restrict__ dst,
                               const float* __restrict__ dinv, const float* __restrict__ h,
                               float* __restrict__ agg) {
  int t = blockIdx.x * blockDim.x + threadIdx.x;
  if (t < N_EDGES * F) {
    int e = t / F, f = t - e * F;
    int s = src[e], d = dst[e];
    float nrm = dinv[s] * dinv[d];
    atomicAdd(&agg[d * F + f], h[s * F + f] * nrm);
  }
}

// ---- tiny dense layers: out = act(a @ W + b) -------------------------------
template <int FI, int FO, bool RELU>
__global__ void k_dense(const float* __restrict__ a, const float* __restrict__ W,
                        const float* __restrict__ b, float* __restrict__ out) {
  int t = blockIdx.x * blockDim.x + threadIdx.x;
  if (t < N_NODES * FO) {
    int i = t / FO, j = t - i * FO;
    float acc = b[j];
#pragma unroll
    for (int k = 0; k < FI; ++k) acc = fmaf(a[i * FI + k], W[k * FO + j], acc);
    out[t] = RELU ? fmaxf(acc, 0.0f) : acc;
  }
}

// ---- final GEMM: out[8192,8192] = A[8192,16] @ B[16,8192] + bias ----------
// One wave per 16x16 tile; K=16 as four chained V_WMMA_F32_16X16X4_F32.
__global__ __launch_bounds__(256) void k_gemm16_wmma(
    const float* __restrict__ A, const float* __restrict__ B,
    const float* __restrict__ bias, float* __restrict__ out) {
  int wave = (blockIdx.x * 256 + (int)threadIdx.x) >> 5;
  int lane = threadIdx.x & 31;
  const int TJ = N_NODES / 16;
  int ti = wave / TJ;
  int tj = wave - ti * TJ;
  int m = lane & 15;       // row within tile (A) / col within tile (B,C,D)
  int half = lane >> 4;    // lane group 0..1
  int row0 = ti * 16;
  int col = tj * 16 + m;

  v8f c = {0.f, 0.f, 0.f, 0.f, 0.f, 0.f, 0.f, 0.f};
#pragma unroll
  for (int s = 0; s < 4; ++s) {
    int k0 = s * 4 + 2 * half;
    // A 16x4 f32 layout: VGPR v, lanes[half] -> A[m][2*half + v]
    const float* ap = A + (row0 + m) * H2 + k0;
    v2f av;
    av[0] = ap[0];
    av[1] = ap[1];
    // B 4x16 f32 layout (dual of A): VGPR v -> B[2*half + v][n]
    v2f bv;
    bv[0] = B[(k0 + 0) * N_NODES + col];
    bv[1] = B[(k0 + 1) * N_NODES + col];
    c = __builtin_amdgcn_wmma_f32_16x16x4_f32(false, av, false, bv, (short)0, c,
                                              false, false);
  }
  float bj = bias[col];
#pragma unroll
  for (int r = 0; r < 8; ++r) {
    int orow = row0 + r + 8 * half;  // C/D layout: VGPR r -> row r + 8*half
    out[orow * N_NODES + col] = c[r] + bj;
  }
}

// ---- driver ----------------------------------------------------------------
extern "C" void kernel_launch(void* const* d_in, const int* in_sizes, int n_in,
                              void* d_out, int out_size, void* d_ws, size_t ws_size,
                              hipStream_t stream) {
  (void)in_sizes; (void)n_in; (void)out_size; (void)ws_size;
  const float* x  = (const float*)d_in[0];
  const int*   ei = (const int*)d_in[1];
  const float* W1 = (const float*)d_in[2];
  const float* b1 = (const float*)d_in[3];
  const float* W2 = (const float*)d_in[4];
  const float* b2 = (const float*)d_in[5];
  const float* W3 = (const float*)d_in[6];
  const float* b3 = (const float*)d_in[7];
  const int* src = ei;            // edge_index[0]
  const int* dst = ei + N_EDGES;  // edge_index[1]

  float* dinv = (float*)d_ws;                      // 8192 floats
  float* bufA = dinv + N_NODES;                    // 8192*32 floats
  float* bufB = bufA + (size_t)N_NODES * H1;       // 8192*32 floats

  const int BT = 256;
  auto blocks = [](long n) { return (int)((n + 255) / 256); };

  // normalization
  k_deg_init<<<blocks(N_NODES), BT, 0, stream>>>(dinv);
  k_deg_count<<<blocks(N_EDGES), BT, 0, stream>>>(dst, dinv);
  k_deg_rsqrt<<<blocks(N_NODES), BT, 0, stream>>>(dinv);

  // layer 1: h1 = relu((Â x) W1 + b1)
  k_agg_init<F_IN><<<blocks((long)N_NODES * F_IN), BT, 0, stream>>>(x, dinv, bufA);
  k_edge_scatter<F_IN><<<blocks((long)N_EDGES * F_IN), BT, 0, stream>>>(src, dst, dinv, x, bufA);
  k_dense<F_IN, H1, true><<<blocks((long)N_NODES * H1), BT, 0, stream>>>(bufA, W1, b1, bufB);

  // layer 2: h2 = relu((Â h1) W2 + b2)
  k_agg_init<H1><<<blocks((long)N_NODES * H1), BT, 0, stream>>>(bufB, dinv, bufA);
  k_edge_scatter<H1><<<blocks((long)N_EDGES * H1), BT, 0, stream>>>(src, dst, dinv, bufB, bufA);
  k_dense<H1, H2, true><<<blocks((long)N_NODES * H2), BT, 0, stream>>>(bufA, W2, b2, bufB);

  // layer 3: out = (Â h2) W3 + b3   (aggregate BEFORE the big GEMM)
  k_agg_init<H2><<<blocks((long)N_NODES * H2), BT, 0, stream>>>(bufB, dinv, bufA);
  k_edge_scatter<H2><<<blocks((long)N_EDGES * H2), BT, 0, stream>>>(src, dst, dinv, bufB, bufA);

  long waves = (long)(N_NODES / 16) * (N_NODES / 16);  // 262144 tiles
  int gblocks = (int)(waves / 8);                      // 8 waves per 256-thread block
  k_gemm16_wmma<<<gblocks, 256, 0, stream>>>(bufA, W3, b3, (float*)d_out);
}